// SemiGCNConv2d_60997125538363
// MI455X (gfx1250) — compile-verified
//
#include <hip/hip_runtime.h>
#include <hip/hip_bf16.h>

// Problem constants from the reference
#define PB     2
#define C_IN   128
#define C_OUT  256
#define NN     16384
#define KK     16

typedef __attribute__((ext_vector_type(2))) float v2f;
typedef __attribute__((ext_vector_type(8))) float v8f;

// -----------------------------------------------------------------------------
// Phase 1: h_t[b][n][o] = relu( sum_i W[o][i] * x[b][i][n] )
// One wave32 computes a 16(n) x 16(o) tile via V_WMMA_F32_16X16X4_F32,
// K-loop over C_IN=128 in steps of 4 (32 WMMAs per tile).
//
// f32 WMMA operand layout (ISA 7.12.2):
//   A (16x4, MxK): lane<16: M=lane,  v0=K0, v1=K1 ; lane>=16: M=lane-16, v0=K2, v1=K3
//   B (4x16, KxN): lane<16: N=lane,  v0=K0, v1=K1 ; lane>=16: N=lane-16, v0=K2, v1=K3
//   C/D (16x16):   lane<16: vr -> M=r, N=lane ; lane>=16: vr -> M=r+8, N=lane-16
// -----------------------------------------------------------------------------
__global__ __launch_bounds__(256) void gemm_relu_wmma(const float* __restrict__ x,
                                                      const float* __restrict__ W,
                                                      float* __restrict__ ht) {
    const int lane = threadIdx.x & 31;
    const int wave = threadIdx.x >> 5;
    const int tile = blockIdx.x * 8 + wave;          // 32768 wave-tiles total

    const int b   = tile / ((NN / 16) * (C_OUT / 16));
    const int rem = tile % ((NN / 16) * (C_OUT / 16));
    const int n0  = (rem / (C_OUT / 16)) * 16;
    const int o0  = (rem % (C_OUT / 16)) * 16;

    const int half = lane >> 4;      // 0: lanes 0-15, 1: lanes 16-31
    const int lm   = lane & 15;      // M (A) / N (B) index within tile

    const float* xb = x + (size_t)b * C_IN * NN;
    const float* wp = W + (size_t)(o0 + lm) * C_IN;  // row of W for this lane's channel

    v8f acc = {};
    #pragma unroll 4
    for (int k0 = 0; k0 < C_IN; k0 += 4) {
        const int ib = k0 + 2 * half;                // this lane's K pair base
        v2f a, bm;
        a.x  = xb[(size_t)ib       * NN + n0 + lm];  // x[b][ib  ][n0+lm]
        a.y  = xb[(size_t)(ib + 1) * NN + n0 + lm];  // x[b][ib+1][n0+lm]
        bm.x = wp[ib];                               // W[o0+lm][ib]
        bm.y = wp[ib + 1];                           // W[o0+lm][ib+1]
        acc = __builtin_amdgcn_wmma_f32_16x16x4_f32(
            /*neg_a=*/false, a, /*neg_b=*/false, bm,
            /*c_mod=*/(short)0, acc, /*reuse_a=*/false, /*reuse_b=*/false);
    }

    // ReLU + store transposed: ht[(b*NN + n)*C_OUT + o], n = n0 + 8*half + r
    float* hp = ht + ((size_t)b * NN + n0 + 8 * half) * C_OUT + o0 + lm;
    #pragma unroll
    for (int r = 0; r < 8; ++r) {
        hp[(size_t)r * C_OUT] = fmaxf(acc[r], 0.0f);
    }
}

// -----------------------------------------------------------------------------
// Phase 2: out[b][o][n] = max_k( ht[b][ idx[b][n][k] ][o] ) * (1/16) + bias[o]
// Block = (b, 32 consecutive n). 256 threads; thread = channel o.
// Gathers are coalesced along o (ht is (B,N,C_OUT), L2-resident: 32 MB << 192 MB).
// Results transposed through padded LDS so the final store is coalesced along n.
// -----------------------------------------------------------------------------
__global__ __launch_bounds__(256) void gather_max_bias(const float* __restrict__ ht,
                                                       const long long* __restrict__ eidx,
                                                       const float* __restrict__ bias,
                                                       float* __restrict__ out) {
    __shared__ int   sidx[32 * KK];          // 512 indices
    __shared__ float sout[C_OUT * 33];       // 33-stride padding: bank-conflict-free

    const int tid   = threadIdx.x;
    const int blk   = blockIdx.x;            // 0..1023
    const int b     = blk >> 9;              // 512 n-tiles per batch
    const int n0    = (blk & 511) * 32;

    // Stage edge_index[0][b][n0 .. n0+31][0..15] (int64 -> int32) in LDS
    const long long* ep = eidx + ((size_t)b * NN + n0) * KK;
    sidx[tid]       = (int)ep[tid];
    sidx[tid + 256] = (int)ep[tid + 256];
    __syncthreads();

    const int   o  = tid;
    const float bo = bias[o];
    const float* hb = ht + (size_t)b * NN * C_OUT + o;

    #pragma unroll 1
    for (int nl = 0; nl < 32; ++nl) {
        const int* ip = &sidx[nl * KK];
        float m = -3.402823466e38f;
        #pragma unroll
        for (int k = 0; k < KK; ++k) {
            const float v = hb[(size_t)ip[k] * C_OUT];   // coalesced across lanes (o)
            m = fmaxf(m, v);
        }
        sout[o * 33 + nl] = m * (1.0f / KK) + bo;
    }
    __syncthreads();

    // Coalesced store: each 32-thread group writes 32 consecutive n for one o
    float* ob = out + (size_t)b * C_OUT * NN + n0;
    #pragma unroll
    for (int f = tid; f < C_OUT * 32; f += 256) {
        const int oo = f >> 5;
        const int nl = f & 31;
        ob[(size_t)oo * NN + nl] = sout[oo * 33 + nl];
    }
}

extern "C" void kernel_launch(void* const* d_in, const int* in_sizes, int n_in,
                              void* d_out, int out_size, void* d_ws, size_t ws_size,
                              hipStream_t stream) {
    const float*     x    = (const float*)d_in[0];     // (B, C_IN, N, 1) f32
    const long long* eidx = (const long long*)d_in[1]; // (2, B, N, K) int64
    const float*     W    = (const float*)d_in[2];     // (C_OUT, C_IN) f32
    const float*     bias = (const float*)d_in[3];     // (1, C_OUT, 1, 1) f32
    float*           out  = (float*)d_out;             // (B, C_OUT, N, 1) f32
    float*           ht   = (float*)d_ws;              // (B, N, C_OUT) f32 = 32 MB

    // Phase 1: WMMA GEMM + ReLU -> transposed h in workspace
    // 2 * 1024 * 16 = 32768 wave-tiles, 8 waves/block -> 4096 blocks
    gemm_relu_wmma<<<4096, 256, 0, stream>>>(x, W, ht);

    // Phase 2: gather + max + scale + bias
    // 2 * 512 n-tiles -> 1024 blocks
    gather_max_bias<<<1024, 256, 0, stream>>>(ht, eidx, bias, out);
}